// MapAnythingTokenBEV_3169685865334
// MI455X (gfx1250) — compile-verified
//
#include <hip/hip_runtime.h>

namespace {
constexpr int B_ = 2, N_ = 6, C_ = 128, H_ = 128, W_ = 352;
constexpr int NXG_ = 256, NYG_ = 256, NZG_ = 1;
constexpr int HW_ = H_ * W_;             // 45056
constexpr int PLANE_ = NYG_ * NXG_;      // 65536
constexpr int NPTS_ = B_ * N_ * H_ * W_; // 540672 = 2112 * 256
}

typedef __attribute__((ext_vector_type(2))) float v2f;
typedef __attribute__((ext_vector_type(8))) float v8f;

__global__ void zero_kernel(float4* __restrict__ out, int n4) {
  int i = blockIdx.x * blockDim.x + threadIdx.x;
  if (i < n4) out[i] = float4{0.f, 0.f, 0.f, 0.f};
}

__global__ __launch_bounds__(256)
void bev_scatter_kernel(const float* __restrict__ feat,
                        const float* __restrict__ depth,
                        const float* __restrict__ rots,
                        const float* __restrict__ trans,
                        const float* __restrict__ intrins,
                        const float* __restrict__ dxp,
                        const float* __restrict__ bxp,
                        float* __restrict__ out) {
  const int lane = threadIdx.x & 31;
  const int wave = threadIdx.x >> 5;
  const int pbase = (blockIdx.x * 8 + wave) * 32;   // 32 consecutive pixels, one row

  const int w0 = pbase % W_;
  int t = pbase / W_;
  const int h = t % H_;  t /= H_;
  const int n = t % N_;
  const int b = t / N_;
  const int cam = b * N_ + n;

  // ---- combine = rots @ inv(intrins); 4th column = trans (wave-uniform) ----
  const float* K = intrins + cam * 9;
  const float k00=K[0],k01=K[1],k02=K[2],
              k10=K[3],k11=K[4],k12=K[5],
              k20=K[6],k21=K[7],k22=K[8];
  const float det = k00*(k11*k22-k12*k21) - k01*(k10*k22-k12*k20) + k02*(k10*k21-k11*k20);
  const float id  = 1.0f / det;
  const float i00 =  (k11*k22-k12*k21)*id, i01 = -(k01*k22-k02*k21)*id, i02 =  (k01*k12-k02*k11)*id;
  const float i10 = -(k10*k22-k12*k20)*id, i11 =  (k00*k22-k02*k20)*id, i12 = -(k00*k12-k02*k10)*id;
  const float i20 =  (k10*k21-k11*k20)*id, i21 = -(k00*k21-k01*k20)*id, i22 =  (k00*k11-k01*k10)*id;
  const float* R = rots + cam * 9;
  const float r00=R[0],r01=R[1],r02=R[2],r10=R[3],r11=R[4],r12=R[5],r20=R[6],r21=R[7],r22=R[8];
  const float m00 = r00*i00+r01*i10+r02*i20, m01 = r00*i01+r01*i11+r02*i21, m02 = r00*i02+r01*i12+r02*i22;
  const float m10 = r10*i00+r11*i10+r12*i20, m11 = r10*i01+r11*i11+r12*i21, m12 = r10*i02+r11*i12+r12*i22;
  const float m20 = r20*i00+r21*i10+r22*i20, m21 = r20*i01+r21*i11+r22*i21, m22 = r20*i02+r21*i12+r22*i22;
  const float* T = trans + cam * 3;
  const float m03 = T[0], m13 = T[1], m23 = T[2];

  // per-lane clamped depth
  float d = depth[cam * HW_ + h * W_ + w0 + lane];
  d = fminf(fmaxf(d, 0.1f), 80.0f);

  // ---- A matrix (16x4 f32): rows 0..2 = [combine | trans], rest zero ----
  // layout: lanes 0-15 hold K=0 (v0) / K=1 (v1); lanes 16-31 hold K=2 / K=3; M = lane&15
  const int m  = lane & 15;
  const int hi = lane >> 4;
  const float a0 = (m==0) ? (hi ? m02 : m00) : (m==1) ? (hi ? m12 : m10) : (m==2) ? (hi ? m22 : m20) : 0.f;
  const float a1 = (m==0) ? (hi ? m03 : m01) : (m==1) ? (hi ? m13 : m11) : (m==2) ? (hi ? m23 : m21) : 0.f;
  const v2f A = {a0, a1};

  const float py = (float)h;
  const v8f cz = {};

  // ---- B matrix (4x16 f32): column n = [x_n*d_n, y_n*d_n, d_n, 1] ----
  // layout: lanes 0-15 hold K=0 (v0) / K=1 (v1); lanes 16-31 hold K=2 / K=3; N = lane&15
  // group 0: wave points 0..15
  const float dq0 = __shfl(d, m, 32);
  const float px0 = (float)(w0 + m);
  const v2f B0 = { hi ? dq0 : px0 * dq0, hi ? 1.0f : py * dq0 };
  const v8f D0 = __builtin_amdgcn_wmma_f32_16x16x4_f32(
      false, A, false, B0, (short)0, cz, false, false);

  // group 1: wave points 16..31
  const float dq1 = __shfl(d, m + 16, 32);
  const float px1 = (float)(w0 + m + 16);
  const v2f B1 = { hi ? dq1 : px1 * dq1, hi ? 1.0f : py * dq1 };
  const v8f D1 = __builtin_amdgcn_wmma_f32_16x16x4_f32(
      false, A, false, B1, (short)0, cz, false, false);

  // D rows 0..2 (x,y,z) live at N=lane for lanes 0-15.
  // Pull group-1 results across with UNCONDITIONAL bpermutes (all lanes execute,
  // no exec-mask branching), then select per lane.
  const float x1 = __shfl(D1[0], m, 32);
  const float y1 = __shfl(D1[1], m, 32);
  const float z1 = __shfl(D1[2], m, 32);
  const float gx = (lane < 16) ? D0[0] : x1;
  const float gy = (lane < 16) ? D0[1] : y1;
  const float gz = (lane < 16) ? D0[2] : z1;

  // voxelize: wave-uniform reciprocals (three divides per wave instead of three
  // full IEEE div chains per lane), truncation-toward-zero matches astype(int32)
  const float dx0 = dxp[0], dx1 = dxp[1], dx2 = dxp[2];
  const float rdx0 = 1.0f / dx0, rdx1 = 1.0f / dx1, rdx2 = 1.0f / dx2;
  const float lo0 = bxp[0] - 0.5f * dx0;
  const float lo1 = bxp[1] - 0.5f * dx1;
  const float lo2 = bxp[2] - 0.5f * dx2;
  const int vx = (int)((gx - lo0) * rdx0);
  const int vy = (int)((gy - lo1) * rdx1);
  const int vz = (int)((gz - lo2) * rdx2);
  const bool valid = (vx >= 0) & (vx < NXG_) & (vy >= 0) & (vy < NYG_) &
                     (vz >= 0) & (vz < NZG_);

  if (valid) {
    const int featbase = cam * (C_ * HW_) + h * W_ + w0 + lane; // c stride = HW_
    const int outbase  = b * (C_ * PLANE_) + vy * NXG_ + vx;    // c stride = PLANE_
#pragma unroll 8
    for (int c = 0; c < C_; ++c) {
      // read-once stream: keep feat out of L2 so the 67MB atomic set stays resident
      const float v = __builtin_nontemporal_load(feat + featbase + c * HW_);
      atomicAdd(out + outbase + c * PLANE_, v);   // global_atomic_add_f32, no return
    }
  }
}

extern "C" void kernel_launch(void* const* d_in, const int* in_sizes, int n_in,
                              void* d_out, int out_size, void* d_ws, size_t ws_size,
                              hipStream_t stream) {
  const float* feat    = (const float*)d_in[0];
  const float* depth   = (const float*)d_in[1];
  const float* rots    = (const float*)d_in[2];
  const float* trans   = (const float*)d_in[3];
  const float* intrins = (const float*)d_in[4];
  const float* dxp     = (const float*)d_in[5];
  const float* bxp     = (const float*)d_in[6];
  float* out = (float*)d_out;

  const int n4 = out_size / 4;  // 16,777,216 floats -> 4,194,304 float4
  zero_kernel<<<(n4 + 255) / 256, 256, 0, stream>>>((float4*)out, n4);

  bev_scatter_kernel<<<NPTS_ / 256, 256, 0, stream>>>(
      feat, depth, rots, trans, intrins, dxp, bxp, out);
}